// TripletMarginLoss_7292854469010
// MI455X (gfx1250) — compile-verified
//
#include <hip/hip_runtime.h>

typedef __attribute__((ext_vector_type(2))) float v2f;
typedef __attribute__((ext_vector_type(8))) float v8f;

#define N_EMB 384
#define D_EMB 128
#define MARGIN_F 1.0f
#define HARD_THRESH_F 1e-16f
#define EPS_COUNT_F 1e-16f

// ---------------------------------------------------------------------------
// Kernel 1: per-row squared norms + zero the global accumulators (fused so the
// stream-ordered zeroing costs no extra launch). 384 rows, 128 threads/block.
// ---------------------------------------------------------------------------
__global__ __launch_bounds__(128)
void sqnorm_init_kernel(const float* __restrict__ X,
                        float* __restrict__ sqn,
                        float* __restrict__ accum) {
    int i = blockIdx.x * blockDim.x + threadIdx.x;
    if (i == 0) {
        accum[0] = 0.0f;   // sum of relu'd triplet losses
        accum[1] = 0.0f;   // n_hard
        accum[2] = 0.0f;   // n_triplets
    }
    if (i < N_EMB) {
        const float4* row = reinterpret_cast<const float4*>(X + i * D_EMB);
        float s = 0.0f;
#pragma unroll
        for (int q = 0; q < D_EMB / 4; ++q) {
            float4 v = row[q];
            s += v.x * v.x + v.y * v.y + v.z * v.z + v.w * v.w;
        }
        sqn[i] = s;
    }
}

// ---------------------------------------------------------------------------
// Kernel 2: G = X * X^T via V_WMMA_F32_16X16X4_F32, one wave per 16x16 tile,
// then distance epilogue d = sqrt(max(sqn_i + sqn_j - 2 g, 0)).
//
// A fragment (16x4 f32, 2 VGPRs): lanes 0-15 hold M=lane, K={kk,kk+1};
//                                 lanes 16-31 hold M=lane-16, K={kk+2,kk+3}.
// B fragment (4x16 f32, 2 VGPRs): mirrored (N in lane, K in VGPR/half-wave);
// since B[k][n] = X[col0+n][k], the per-lane load pattern matches A's.
// C/D (16x16 f32, 8 VGPRs): VGPR r, lane L -> M = r + 8*(L>=16), N = L%16.
// ---------------------------------------------------------------------------
__global__ __launch_bounds__(32)
void dist_wmma_kernel(const float* __restrict__ X,
                      const float* __restrict__ sqn,
                      float* __restrict__ dist) {
    const int row0 = blockIdx.y * 16;
    const int col0 = blockIdx.x * 16;
    const int lane = threadIdx.x;
    const int m    = lane & 15;
    const int koff = (lane >> 4) << 1;      // 0 for lanes 0-15, 2 for 16-31

    const float* arow = X + (row0 + m) * D_EMB;
    const float* brow = X + (col0 + m) * D_EMB;

    v8f c = {};
#pragma unroll
    for (int kk = 0; kk < D_EMB; kk += 4) {
        v2f a = *reinterpret_cast<const v2f*>(arow + kk + koff);
        v2f b = *reinterpret_cast<const v2f*>(brow + kk + koff);
        // (neg_a, A, neg_b, B, c_mod, C, reuse_a, reuse_b)
        c = __builtin_amdgcn_wmma_f32_16x16x4_f32(
                false, a, false, b, (short)0, c, false, false);
    }

    const int rbase = row0 + ((lane >> 4) << 3);
    const int col   = col0 + m;
    const float sc  = sqn[col];
#pragma unroll
    for (int r = 0; r < 8; ++r) {
        const int row = rbase + r;
        float sq = sqn[row] + sc - 2.0f * c[r];
        float dv = (sq > 0.0f) ? sqrtf(sq) : 0.0f;
        dist[row * N_EMB + col] = dv;
    }
}

// ---------------------------------------------------------------------------
// Kernel 3: per-anchor triplet reduction. One block per anchor i.
// valid(i,j,k) = (j != i && label_j == label_i) && (label_k != label_i);
// j != k and i != k are implied by the label conditions.
// Distance row + labels staged in LDS; inner k-loop reads only LDS.
// ---------------------------------------------------------------------------
__global__ __launch_bounds__(256)
void triplet_kernel(const float* __restrict__ dist,
                    const int* __restrict__ labels,
                    float* __restrict__ accum) {
    __shared__ float srow[N_EMB];
    __shared__ int   slab[N_EMB];
    __shared__ float red[3];

    const int i   = blockIdx.x;
    const int tid = threadIdx.x;

    for (int t = tid; t < N_EMB; t += 256) {
        srow[t] = dist[i * N_EMB + t];
        slab[t] = labels[t];
    }
    if (tid == 0) { red[0] = 0.0f; red[1] = 0.0f; red[2] = 0.0f; }
    __syncthreads();

    const int la = slab[i];
    float s    = 0.0f;
    float hard = 0.0f;
    float cnt  = 0.0f;

    for (int j = 0; j < N_EMB; ++j) {
        if (j == i || slab[j] != la) continue;      // uniform branch (LDS data)
        const float dij = srow[j];
        for (int k = tid; k < N_EMB; k += 256) {
            if (slab[k] != la) {
                const float v = dij - srow[k] + MARGIN_F;
                if (v > 0.0f)          s += v;        // relu'd loss
                if (v > HARD_THRESH_F) hard += 1.0f;  // hard count
                cnt += 1.0f;                          // valid-triplet count
            }
        }
    }

    atomicAdd(&red[0], s);
    atomicAdd(&red[1], hard);
    atomicAdd(&red[2], cnt);
    __syncthreads();
    if (tid == 0) {
        atomicAdd(&accum[0], red[0]);
        atomicAdd(&accum[1], red[1]);
        atomicAdd(&accum[2], red[2]);
    }
}

// ---------------------------------------------------------------------------
// Kernel 4: finalize scalars.  out[0] = loss, out[1] = fraction_hard.
// ---------------------------------------------------------------------------
__global__ void finalize_kernel(const float* __restrict__ accum,
                                float* __restrict__ out) {
    out[0] = accum[0] / (accum[1] + EPS_COUNT_F);
    out[1] = accum[1] / (accum[2] + EPS_COUNT_F);
}

extern "C" void kernel_launch(void* const* d_in, const int* in_sizes, int n_in,
                              void* d_out, int out_size, void* d_ws, size_t ws_size,
                              hipStream_t stream) {
    const float* X      = (const float*)d_in[0];   // [384,128] f32
    const int*   labels = (const int*)d_in[1];     // [384] int
    float*       out    = (float*)d_out;           // [2] f32: loss, fraction_hard

    float* ws    = (float*)d_ws;
    float* sqn   = ws;                              // 384 floats
    float* dist  = ws + N_EMB;                      // 384*384 floats
    float* accum = ws + N_EMB + N_EMB * N_EMB;      // 3 floats

    sqnorm_init_kernel<<<(N_EMB + 127) / 128, 128, 0, stream>>>(X, sqn, accum);

    dim3 grid(N_EMB / 16, N_EMB / 16);
    dist_wmma_kernel<<<grid, 32, 0, stream>>>(X, sqn, dist);

    triplet_kernel<<<N_EMB, 256, 0, stream>>>(dist, labels, accum);

    finalize_kernel<<<1, 1, 0, stream>>>(accum, out);
}